// CenterDirGroundtruth_17970143166672
// MI455X (gfx1250) — compile-verified
//
#include <hip/hip_runtime.h>
#include <math.h>

// ---------------------------------------------------------------------------
// CenterDirGroundtruth for MI455X (gfx1250, wave32).
//
// Roofline: ~50 MB of traffic -> ~2-4 us at 23.3 TB/s; compute is negligible.
// Strategy: 4 kernels, separable Gaussian blur (row sums of the 2D kernel
// give the exact normalized 1D kernel), each 1D conv pass expressed as a
// 16x36 x 36x16 banded matmul on V_WMMA_F32_16X16X4_F32 (9 K-steps of 4),
// one wave per 16x16 tile, tiles staged through LDS.
//
// Round-1 fix: band matrix lookups go through a zero-padded LDS table
// (w1pad[64], index k-n+15 in [0,50]) so operand fetch is branch-free and
// each chunk's band operand is one ds_load_b64; LDS strides are 41 (odd,
// coprime with 64 banks) for conflict-free half-wave reads.
// ---------------------------------------------------------------------------

#define BB   4
#define NC   64
#define HH   512
#define WW   512
#define HWP  (HH * WW)
#define KS   21
#define PADK 10

typedef __attribute__((ext_vector_type(2))) float v2f;
typedef __attribute__((ext_vector_type(8))) float v8f;

__device__ __forceinline__ int reflect_idx(int i, int n) {
    // numpy 'reflect' (no edge duplication); |overhang| <= PADK, single fold ok
    if (i < 0) i = -i;
    if (i >= n) i = 2 * n - 2 - i;
    return i;
}

// Build the zero-padded 1D kernel table: w1pad[i] = w1[i-15] for i in [15,35],
// else 0.  w1[d] = sum_j k2d[d][j] (row sums recover the normalized 1D kernel).
__device__ __forceinline__ void build_w1pad(const float* __restrict__ k2d,
                                            float* __restrict__ w1pad) {
    if (threadIdx.x < 64) {
        int d = (int)threadIdx.x - 15;
        float v = 0.0f;
        if (d >= 0 && d < KS) {
            for (int j = 0; j < KS; ++j) v += k2d[d * KS + j];
        }
        w1pad[threadIdx.x] = v;
    }
}

// ---------------------------------------------------------------------------
// Kernel 1: per-pixel nearest-center fields (channels 0..4) + binary mask0.
// ---------------------------------------------------------------------------
__global__ void fields_kernel(const float* __restrict__ centers,
                              const float* __restrict__ gt_in,
                              float* __restrict__ out,
                              float* __restrict__ mask0,
                              unsigned* __restrict__ maxes) {
    __shared__ float scy[NC];
    __shared__ float scx[NC];

    int p   = blockIdx.x * blockDim.x + threadIdx.x;
    int b   = p >> 18;            // HW = 2^18
    int rem = p & (HWP - 1);
    int y   = rem >> 9;           // row
    int x   = rem & (WW - 1);     // col

    if (threadIdx.x < NC) {
        scy[threadIdx.x] = centers[(b * NC + threadIdx.x) * 2 + 0];
        scx[threadIdx.x] = centers[(b * NC + threadIdx.x) * 2 + 1];
    }
    if (p < BB) maxes[p] = 0u;    // clear per-batch max slots (ws not re-poisoned)
    __syncthreads();

    float fy = (float)y, fx = (float)x;
    float best = __builtin_inff();
    int   bid  = 0;
    bool  anyValid = false;
    for (int j = 0; j < NC; ++j) {
        float cy = scy[j], cx = scx[j];
        bool  vj = (cy > 0.0f) || (cx > 0.0f);
        anyValid = anyValid || vj;
        float dx = fy - cx;           // grid_h - cx  (reference's row axis)
        float dy = fx - cy;           // grid_w - cy
        float d2 = dx * dx + dy * dy;
        d2 = vj ? d2 : __builtin_inff();
        if (d2 < best) { best = d2; bid = j; }   // strict '<' == argmin first index
    }

    float gtX = scx[bid] - fy;
    float gtY = scy[bid] - fx;
    float r2  = gtX * gtX + gtY * gtY;
    float gtR = (r2 > 0.0f) ? sqrtf(r2) : 0.0f;
    float th  = atan2f(gtY, (r2 > 0.0f) ? gtX : 1.0f);
    float s   = sinf(th);
    float co  = cosf(th);
    float M   = sqrtf(s * s + co * co);
    s /= M; co /= M;

    bool near = (fabsf(gtX) < 3.0f) && (fabsf(gtY) < 3.0f);
    mask0[(size_t)b * HWP + rem] = (anyValid && near) ? 1.0f : 0.0f;

    size_t base = (size_t)b * 6 * HWP + rem;
    if (anyValid) {
        out[base + 0 * HWP] = gtR;
        out[base + 1 * HWP] = th;
        out[base + 2 * HWP] = s;
        out[base + 3 * HWP] = co;
    } else {
        out[base + 0 * HWP] = gt_in[base + 0 * HWP];
        out[base + 1 * HWP] = gt_in[base + 1 * HWP];
        out[base + 2 * HWP] = gt_in[base + 2 * HWP];
        out[base + 3 * HWP] = gt_in[base + 3 * HWP];
    }
    out[base + 4 * HWP] = gt_in[base + 4 * HWP];
}

// ---------------------------------------------------------------------------
// Kernel 2: horizontal blur pass as WMMA.  Out(16x16) = A(16x36) x Band(36x16).
// One wave per 16x16 tile, 4 waves per block.
// ---------------------------------------------------------------------------
__global__ void hblur_wmma(const float* __restrict__ mask0,
                           const float* __restrict__ k2d,
                           float* __restrict__ tmp) {
    __shared__ float w1pad[64];
    __shared__ float tA[4][16][41];   // 36 cols used, stride 41 (bank-free)

    int lane = threadIdx.x & 31;
    int wv   = threadIdx.x >> 5;

    build_w1pad(k2d, w1pad);

    int tile = blockIdx.x * 4 + wv;   // 4096 tiles total
    int b    = tile >> 10;
    int rem  = tile & 1023;
    int r0   = (rem >> 5) << 4;
    int c0   = (rem & 31) << 4;

    const float* img = mask0 + (size_t)b * HWP;
    for (int i = lane; i < 16 * 36; i += 32) {     // uniform 18 iterations
        int r = i / 36, m = i - r * 36;
        tA[wv][r][m] = img[(r0 + r) * WW + reflect_idx(c0 - PADK + m, WW)];
    }
    __syncthreads();

    int off = (lane >> 4) * 2;        // f32 A-matrix 16x4 layout: K = k0+off, k0+off+1
    int nr  = lane & 15;
    const float* arow  = &tA[wv][nr][off];        // A tile row, consecutive pairs
    const float* bandp = &w1pad[off - nr + 15];   // band col, consecutive pairs
    v8f acc = {};
    for (int kc = 0; kc < 9; ++kc) {  // K = 36 in steps of 4
        int k0 = kc * 4;
        v2f a, bb;
        a.x  = arow[k0];
        a.y  = arow[k0 + 1];
        bb.x = bandp[k0];
        bb.y = bandp[k0 + 1];
        acc = __builtin_amdgcn_wmma_f32_16x16x4_f32(
            false, a, false, bb, (short)0, acc, false, false);
    }

    float* dst = tmp + (size_t)b * HWP;
    int mbase = (lane >> 4) * 8;      // C/D layout: lanes 16-31 hold rows M+8
    for (int v = 0; v < 8; ++v)
        dst[(r0 + mbase + v) * WW + c0 + nr] = acc[v];
}

// ---------------------------------------------------------------------------
// Kernel 3: vertical blur pass as WMMA.  Out(16x16) = Band(16x36) x Tile(36x16).
// Tile staged TRANSPOSED in LDS so the B operand is consecutive pairs too.
// Writes channel 5 of d_out (unnormalized) + per-batch max via atomicMax.
// ---------------------------------------------------------------------------
__global__ void vblur_wmma(const float* __restrict__ tmp,
                           const float* __restrict__ k2d,
                           float* __restrict__ out,
                           unsigned* __restrict__ maxes) {
    __shared__ float w1pad[64];
    __shared__ float tBt[4][16][41];  // tBt[n][m] = tile[m][n]; stride 41

    int lane = threadIdx.x & 31;
    int wv   = threadIdx.x >> 5;

    build_w1pad(k2d, w1pad);

    int tile = blockIdx.x * 4 + wv;
    int b    = tile >> 10;
    int rem  = tile & 1023;
    int r0   = (rem >> 5) << 4;
    int c0   = (rem & 31) << 4;

    const float* src = tmp + (size_t)b * HWP;
    for (int i = lane; i < 36 * 16; i += 32) {     // uniform 18 iterations
        int m = i >> 4, n = i & 15;                // n fastest: coalesced global
        tBt[wv][n][m] = src[reflect_idx(r0 - PADK + m, HH) * WW + c0 + n];
    }
    __syncthreads();

    int off = (lane >> 4) * 2;
    int nr  = lane & 15;
    const float* bandp = &w1pad[off - nr + 15];   // A = transposed band matrix
    const float* bcol  = &tBt[wv][nr][off];       // B = tile column, consecutive
    v8f acc = {};
    for (int kc = 0; kc < 9; ++kc) {
        int k0 = kc * 4;
        v2f a, bb;
        a.x  = bandp[k0];
        a.y  = bandp[k0 + 1];
        bb.x = bcol[k0];
        bb.y = bcol[k0 + 1];
        acc = __builtin_amdgcn_wmma_f32_16x16x4_f32(
            false, a, false, bb, (short)0, acc, false, false);
    }

    float* dst = out + ((size_t)b * 6 + 5) * HWP;
    int mbase = (lane >> 4) * 8;
    float mx = 0.0f;
    for (int v = 0; v < 8; ++v) {
        float val = acc[v];
        dst[(r0 + mbase + v) * WW + c0 + nr] = val;
        mx = fmaxf(mx, val);
    }
    // wave32 cross-lane max reduce, then one atomic per wave
    for (int d = 16; d > 0; d >>= 1)
        mx = fmaxf(mx, __shfl_xor(mx, d, 32));
    if (lane == 0)
        atomicMax(&maxes[b], __float_as_uint(mx));   // nonneg f32: uint order ok
}

// ---------------------------------------------------------------------------
// Kernel 4: normalize channel 5 in place by the per-batch max.
// ---------------------------------------------------------------------------
__global__ void normalize_kernel(float* __restrict__ out,
                                 const unsigned* __restrict__ maxes) {
    int p   = blockIdx.x * blockDim.x + threadIdx.x;
    int b   = p >> 18;
    int rem = p & (HWP - 1);
    float m = __uint_as_float(maxes[b]);
    size_t idx = ((size_t)b * 6 + 5) * HWP + rem;
    float v = out[idx];
    out[idx] = (m > 0.0f) ? v / fmaxf(m, 1e-12f) : 0.0f;
}

// ---------------------------------------------------------------------------
extern "C" void kernel_launch(void* const* d_in, const int* in_sizes, int n_in,
                              void* d_out, int out_size, void* d_ws, size_t ws_size,
                              hipStream_t stream) {
    const float* centers = (const float*)d_in[0];   // (4,64,2)
    const float* gt_in   = (const float*)d_in[1];   // (4,6,512,512)
    const float* k2d     = (const float*)d_in[2];   // (21,21)
    float*       out     = (float*)d_out;

    char* ws = (char*)d_ws;
    float*    mask0 = (float*)(ws);                           // 4 MiB
    float*    tmp   = (float*)(ws + (size_t)BB * HWP * 4);    // 4 MiB
    unsigned* maxes = (unsigned*)(ws + (size_t)2 * BB * HWP * 4);

    // 1) fields + mask0 (+ clear max slots)
    fields_kernel<<<(BB * HWP) / 256, 256, 0, stream>>>(centers, gt_in, out, mask0, maxes);
    // 2) horizontal Gaussian pass (WMMA), 4096 tiles / 4 waves per block
    hblur_wmma<<<1024, 128, 0, stream>>>(mask0, k2d, tmp);
    // 3) vertical Gaussian pass (WMMA) -> channel 5 + per-batch max
    vblur_wmma<<<1024, 128, 0, stream>>>(tmp, k2d, out, maxes);
    // 4) normalize channel 5
    normalize_kernel<<<(BB * HWP) / 256, 256, 0, stream>>>(out, maxes);
}